// PrattRNNLayer_80221399154928
// MI455X (gfx1250) — compile-verified
//
#include <hip/hip_runtime.h>
#include <hip/hip_bf16.h>
#include <math.h>

typedef __attribute__((ext_vector_type(2))) float v2f;
typedef __attribute__((ext_vector_type(8))) float v8f;

#define DDIM 128
#define TLEN 2048
#define BATCH 128
#define LPAD 4   // LDS row padding (floats) to break bank conflicts

// ---------------------------------------------------------------------------
// Kernel 1: d_out[m, e] = cos( x[m, :] @ kernel[:, e] )^2   for m in [0, B*T)
// Fully parallel GEMM using V_WMMA_F32_16X16X4_F32.
// Each block: 8 waves, wave j owns M-tile (blockIdx*8 + j) = 16 rows, all 128
// output columns (8 N-tiles, 64 accumulator VGPRs). kernel matrix staged
// transposed in LDS so B fragments are contiguous b64 reads.
// ---------------------------------------------------------------------------
__global__ void __launch_bounds__(256) hopf_gemm_cos2(
    const float* __restrict__ x,
    const float* __restrict__ kern,
    float* __restrict__ out) {
  __shared__ float Kt[DDIM][DDIM + LPAD];  // Kt[e][d] = kernel[d][e]

  const int tid = threadIdx.x;
  for (int i = tid; i < DDIM * DDIM; i += 256) {
    const int d = i >> 7;
    const int e = i & 127;
    Kt[e][d] = kern[i];
  }
  __syncthreads();

  const int wave = tid >> 5;
  const int lane = tid & 31;
  const int half = lane >> 4;   // 0: K pair {0,1}; 1: K pair {2,3}
  const int lr   = lane & 15;

  const long mtile = (long)blockIdx.x * 8 + wave;
  const long m0    = mtile * 16;
  const float* arow = x + (m0 + lr) * DDIM + 2 * half;

  v8f c[8];
#pragma unroll
  for (int n = 0; n < 8; ++n) c[n] = v8f{};

  for (int k0 = 0; k0 < DDIM; k0 += 4) {
    const v2f a = *(const v2f*)(arow + k0);          // A[lr][k0+2h .. +1]
#pragma unroll
    for (int n = 0; n < 8; ++n) {
      const int col = n * 16 + lr;
      const v2f b = *(const v2f*)(&Kt[col][k0 + 2 * half]);  // B[k][col]
      c[n] = __builtin_amdgcn_wmma_f32_16x16x4_f32(
          false, a, false, b, (short)0, c[n], false, false);
    }
  }

  // Epilogue: cos^2 and store. C layout: VGPR v -> row (v + 8*half), col lr.
#pragma unroll
  for (int n = 0; n < 8; ++n) {
    const int col = n * 16 + lr;
#pragma unroll
    for (int v = 0; v < 8; ++v) {
      const long m = m0 + v + 8 * half;
      const float cv = cosf(c[n][v]);
      out[m * DDIM + col] = cv * cv;
    }
  }
}

// ---------------------------------------------------------------------------
// Kernel 2: sequential scan over T. In-place on d_out (which holds cos^2(km)).
//   rec   = S @ R                      (fp32 WMMA, 32 ops/wave/step)
//   ns    = sqrt(c2 + sin(rec)^2)
//   out   = ns * (1 - ns^2)            (Hopf normal form, mu = 1)
//   S    <- ns
// 8 blocks (16 batch rows each) x 8 waves (16 cols each).
// ---------------------------------------------------------------------------
__global__ void __launch_bounds__(256) hopf_scan(
    const float* __restrict__ rker,
    float* __restrict__ out) {
  __shared__ float Rt[DDIM][DDIM + LPAD];  // Rt[e][d] = R[d][e]
  __shared__ float S[16][DDIM + LPAD];     // state tile [batch_row][dim]

  const int tid = threadIdx.x;
  for (int i = tid; i < DDIM * DDIM; i += 256) {
    const int d = i >> 7;
    const int e = i & 127;
    Rt[e][d] = rker[i];
  }
  for (int i = tid; i < 16 * DDIM; i += 256) {
    S[i >> 7][i & 127] = 0.0f;             // s0 = 0
  }
  __syncthreads();

  const int wave = tid >> 5;
  const int lane = tid & 31;
  const int half = lane >> 4;
  const int lr   = lane & 15;

  const int  b0 = blockIdx.x * 16;         // batch-row tile
  const int  n0 = wave * 16;               // column tile owned by this wave
  float* obase = out + (long)b0 * TLEN * DDIM + n0 + lr;

  for (int t = 0; t < TLEN; ++t) {
    v8f acc = v8f{};
#pragma unroll
    for (int k0 = 0; k0 < DDIM; k0 += 4) {
      const v2f a = *(const v2f*)(&S[lr][k0 + 2 * half]);          // S[row][k]
      const v2f b = *(const v2f*)(&Rt[n0 + lr][k0 + 2 * half]);    // R[k][col]
      acc = __builtin_amdgcn_wmma_f32_16x16x4_f32(
          false, a, false, b, (short)0, acc, false, false);
    }

    float ns[8];
#pragma unroll
    for (int v = 0; v < 8; ++v) {
      const int  row = v + 8 * half;
      float* p = obase + (long)row * TLEN * DDIM + (long)t * DDIM;
      const float c2  = *p;                 // cos^2(km) from kernel 1
      const float sv  = sinf(acc[v]);
      const float ns2 = c2 + sv * sv;       // new_state^2
      const float n_  = sqrtf(ns2);
      ns[v] = n_;
      *p = n_ * (1.0f - ns2);               // hopf_bifur(ns, ns^2)
    }

    __syncthreads();                        // all waves done reading S[t]
#pragma unroll
    for (int v = 0; v < 8; ++v) {
      S[v + 8 * half][n0 + lr] = ns[v];
    }
    __syncthreads();                        // S[t+1] visible to all waves
  }
}

// ---------------------------------------------------------------------------
extern "C" void kernel_launch(void* const* d_in, const int* in_sizes, int n_in,
                              void* d_out, int out_size, void* d_ws, size_t ws_size,
                              hipStream_t stream) {
  (void)in_sizes; (void)n_in; (void)out_size; (void)d_ws; (void)ws_size;
  const float* x    = (const float*)d_in[0];
  const float* kern = (const float*)d_in[1];
  const float* rker = (const float*)d_in[2];
  float* out = (float*)d_out;

  // Phase 1: cos^2(x @ kernel) -> d_out.  M-tiles = B*T/16 = 16384, 8/block.
  hopf_gemm_cos2<<<(BATCH * TLEN) / (16 * 8), 256, 0, stream>>>(x, kern, out);
  // Phase 2: sequential recurrence, in place on d_out.
  hopf_scan<<<BATCH / 16, 256, 0, stream>>>(rker, out);
}